// PointNetSetAbstraction_66889820668134
// MI455X (gfx1250) — compile-verified
//
#include <hip/hip_runtime.h>

#define B_      16
#define N_      4096
#define NPOINT_ 1024
#define NSAMPLE_ 32
#define P_TOT   (B_ * NPOINT_ * NSAMPLE_)   // 524288 positions
#define RAD2    0.04f
#define BN_EPS  1e-5f

typedef __attribute__((ext_vector_type(2))) float v2f;
typedef __attribute__((ext_vector_type(8))) float v8f;

// ---------------------------------------------------------------------------
// 1) Farthest point sampling: one workgroup (1024 threads = 32 waves) per batch.
// ---------------------------------------------------------------------------
__global__ __launch_bounds__(1024) void fps_kernel(const float* __restrict__ xyz,
                                                   float* __restrict__ new_xyz) {
  __shared__ float dist[N_];
  __shared__ float wv[32];
  __shared__ int   wi[32];
  __shared__ float scx, scy, scz;
  __shared__ int   sfar;
  const int b    = blockIdx.x;
  const int tid  = threadIdx.x;
  const int lane = tid & 31;
  const int wid  = tid >> 5;
  const float* xb = xyz + (size_t)b * N_ * 3;

  for (int p = tid; p < N_; p += 1024) dist[p] = 1e10f;
  if (tid == 0) sfar = 0;
  __syncthreads();

  for (int it = 0; it < NPOINT_; ++it) {
    if (tid == 0) {
      int f = sfar;
      float x = xb[f * 3 + 0], y = xb[f * 3 + 1], z = xb[f * 3 + 2];
      scx = x; scy = y; scz = z;
      float* o = new_xyz + ((size_t)b * NPOINT_ + it) * 3;
      o[0] = x; o[1] = y; o[2] = z;
    }
    __syncthreads();
    const float cx = scx, cy = scy, cz = scz;

    float bestv = -1.0f;
    int   besti = 0;
#pragma unroll
    for (int j = 0; j < 4; ++j) {
      int p = tid + j * 1024;
      float dx = xb[p * 3 + 0] - cx;
      float dy = xb[p * 3 + 1] - cy;
      float dz = xb[p * 3 + 2] - cz;
      float d  = dx * dx + dy * dy + dz * dz;
      float nd = fminf(dist[p], d);
      dist[p]  = nd;
      if (nd > bestv) { bestv = nd; besti = p; }
    }
#pragma unroll
    for (int off = 16; off > 0; off >>= 1) {
      float ov = __shfl_down(bestv, off, 32);
      int   oi = __shfl_down(besti, off, 32);
      if (ov > bestv || (ov == bestv && oi < besti)) { bestv = ov; besti = oi; }
    }
    if (lane == 0) { wv[wid] = bestv; wi[wid] = besti; }
    __syncthreads();
    if (wid == 0) {
      bestv = wv[lane];
      besti = wi[lane];
#pragma unroll
      for (int off = 16; off > 0; off >>= 1) {
        float ov = __shfl_down(bestv, off, 32);
        int   oi = __shfl_down(besti, off, 32);
        if (ov > bestv || (ov == bestv && oi < besti)) { bestv = ov; besti = oi; }
      }
      if (lane == 0) sfar = besti;
    }
    __syncthreads();
  }
}

// ---------------------------------------------------------------------------
// 2) Ball query + grouping. One wave per centroid, ballot/prefix-popcount
//    reproduces "first NSAMPLE indices in ascending index order".
// ---------------------------------------------------------------------------
__global__ __launch_bounds__(256) void group_kernel(const float* __restrict__ xyz,
                                                    const float* __restrict__ pts,
                                                    const float* __restrict__ new_xyz,
                                                    float* __restrict__ x0) {
  const int wid  = blockIdx.x * 8 + (threadIdx.x >> 5);
  const int lane = threadIdx.x & 31;
  const int b = wid >> 10;
  const int s = wid & 1023;
  const float* xb = xyz + (size_t)b * N_ * 3;
  const float* pb = pts + (size_t)b * N_ * 3;
  const float* c  = new_xyz + ((size_t)b * NPOINT_ + s) * 3;
  const float cx = c[0], cy = c[1], cz = c[2];
  const size_t pos0 = ((size_t)b * NPOINT_ + s) * NSAMPLE_;

  int cnt = 0;
  int firstIdx = -1;
  for (int base = 0; base < N_ && cnt < NSAMPLE_; base += 32) {
    const int p = base + lane;
    float dx = xb[p * 3 + 0] - cx;
    float dy = xb[p * 3 + 1] - cy;
    float dz = xb[p * 3 + 2] - cz;
    bool pred = (dx * dx + dy * dy + dz * dz) <= RAD2;
    unsigned m = (unsigned)__ballot(pred);
    if (firstIdx < 0 && m) firstIdx = base + (__ffs((int)m) - 1);
    if (pred) {
      int slot = cnt + __popc(m & ((1u << lane) - 1u));
      if (slot < NSAMPLE_) {
        size_t pos = pos0 + slot;
        x0[0 * (size_t)P_TOT + pos] = dx;
        x0[1 * (size_t)P_TOT + pos] = dy;
        x0[2 * (size_t)P_TOT + pos] = dz;
        x0[3 * (size_t)P_TOT + pos] = pb[p * 3 + 0];
        x0[4 * (size_t)P_TOT + pos] = pb[p * 3 + 1];
        x0[5 * (size_t)P_TOT + pos] = pb[p * 3 + 2];
        x0[6 * (size_t)P_TOT + pos] = 0.f;
        x0[7 * (size_t)P_TOT + pos] = 0.f;
      }
    }
    cnt += __popc(m);
  }
  const int fillFrom = (cnt < NSAMPLE_) ? cnt : NSAMPLE_;
  if (lane >= fillFrom && lane < NSAMPLE_) {
    const int p = firstIdx;
    size_t pos = pos0 + lane;
    float dx = xb[p * 3 + 0] - cx;
    float dy = xb[p * 3 + 1] - cy;
    float dz = xb[p * 3 + 2] - cz;
    x0[0 * (size_t)P_TOT + pos] = dx;
    x0[1 * (size_t)P_TOT + pos] = dy;
    x0[2 * (size_t)P_TOT + pos] = dz;
    x0[3 * (size_t)P_TOT + pos] = pb[p * 3 + 0];
    x0[4 * (size_t)P_TOT + pos] = pb[p * 3 + 1];
    x0[5 * (size_t)P_TOT + pos] = pb[p * 3 + 2];
    x0[6 * (size_t)P_TOT + pos] = 0.f;
    x0[7 * (size_t)P_TOT + pos] = 0.f;
  }
}

// ---------------------------------------------------------------------------
// Zero-pad W(cout x cinR) -> Wpad(cout x CINP). Removes all lane-divergent
// guards from the GEMM inner loop.
// ---------------------------------------------------------------------------
__global__ __launch_bounds__(256) void pad_w_kernel(const float* __restrict__ W,
                                                    float* __restrict__ Wpad,
                                                    int cinR, int cinP, int n) {
  int i = blockIdx.x * 256 + threadIdx.x;  // over cout*cinP
  if (i < n) {
    int r = i / cinP, k = i % cinP;
    Wpad[i] = (k < cinR) ? W[r * cinR + k] : 0.f;
  }
}

// ---------------------------------------------------------------------------
// 3) fp32 WMMA GEMM: Y(cout x P) = Wpad(cout x CINP) * X(CINP x P) + bias.
//    One wave per 16(cout) x 32(positions) tile: two independent accumulators
//    share each A fragment; fully unrolled K-loop of V_WMMA_F32_16X16X4_F32.
//    Grid divides exactly -> every wave fully active (EXEC all ones for WMMA).
//    Operand layouts per ISA:
//      A 16x4: lanes 0-15 rows M; VGPR0/1 hold K {0,1} (half 0) or {2,3} (half 1)
//      B 4x16: VGPR0 rows {k0,k0+2} split by lane half, VGPR1 rows {k0+1,k0+3}
//      C/D 16x16: VGPR i -> rows i (lanes 0-15) / i+8 (lanes 16-31)
// ---------------------------------------------------------------------------
template <int CINP>
__global__ __launch_bounds__(256) void wmma_gemm_kernel(const float* __restrict__ Wp,
                                                        const float* __restrict__ bias,
                                                        const float* __restrict__ X,
                                                        float* __restrict__ Y,
                                                        int cout) {
  const int gw   = blockIdx.x * 8 + (threadIdx.x >> 5);
  const int lane = threadIdx.x & 31;
  const int mt   = cout >> 4;
  const int m0   = (gw % mt) << 4;
  const int n0   = (gw / mt) << 5;   // 32 positions per wave (2 N-tiles)
  const int half = lane >> 4;
  const int l15  = lane & 15;
  const int m    = m0 + l15;
  const int n    = n0 + l15;

  v8f acc0 = {};
  v8f acc1 = {};
#pragma unroll
  for (int k0 = 0; k0 < CINP; k0 += 4) {
    const int kk = k0 + half * 2;
    v2f a, b0v, b1v;
    a.x   = Wp[m * CINP + kk];
    a.y   = Wp[m * CINP + kk + 1];
    b0v.x = X[(size_t)kk       * P_TOT + n];
    b0v.y = X[(size_t)(kk + 1) * P_TOT + n];
    b1v.x = X[(size_t)kk       * P_TOT + n + 16];
    b1v.y = X[(size_t)(kk + 1) * P_TOT + n + 16];
    acc0 = __builtin_amdgcn_wmma_f32_16x16x4_f32(false, a, false, b0v,
                                                 (short)0, acc0, false, false);
    acc1 = __builtin_amdgcn_wmma_f32_16x16x4_f32(false, a, false, b1v,
                                                 (short)0, acc1, false, false);
  }
#pragma unroll
  for (int i = 0; i < 8; ++i) {
    const int row = m0 + i + half * 8;
    const float bb = bias[row];
    Y[(size_t)row * P_TOT + n]      = acc0[i] + bb;
    Y[(size_t)row * P_TOT + n + 16] = acc1[i] + bb;
  }
}

// ---------------------------------------------------------------------------
// 4) Per-channel sum / sum-of-squares (BN batch statistics over B,K,S).
// ---------------------------------------------------------------------------
__global__ __launch_bounds__(256) void zero_stats_kernel(float* __restrict__ s, int n) {
  int i = blockIdx.x * 256 + threadIdx.x;
  if (i < n) s[i] = 0.f;
}

__global__ __launch_bounds__(256) void stats_kernel(const float* __restrict__ Y,
                                                    float* __restrict__ ssum,
                                                    float* __restrict__ ssq,
                                                    int chunks) {
  const int c  = blockIdx.x / chunks;
  const int ch = blockIdx.x % chunks;
  const int epc = P_TOT / chunks;
  const float* y = Y + (size_t)c * P_TOT + (size_t)ch * epc;
  float ls = 0.f, lq = 0.f;
  for (int i = threadIdx.x; i < epc; i += 256) {
    float v = y[i];
    ls += v;
    lq += v * v;
  }
#pragma unroll
  for (int off = 16; off > 0; off >>= 1) {
    ls += __shfl_down(ls, off, 32);
    lq += __shfl_down(lq, off, 32);
  }
  __shared__ float sws[8], swq[8];
  const int lane = threadIdx.x & 31, wid = threadIdx.x >> 5;
  if (lane == 0) { sws[wid] = ls; swq[wid] = lq; }
  __syncthreads();
  if (threadIdx.x == 0) {
    float ts = 0.f, tq = 0.f;
#pragma unroll
    for (int i = 0; i < 8; ++i) { ts += sws[i]; tq += swq[i]; }
    atomicAdd(&ssum[c], ts);
    atomicAdd(&ssq[c], tq);
  }
}

// ---------------------------------------------------------------------------
// 5) BN (from batch stats) + ReLU, in place. grid = (P/256, cout).
// ---------------------------------------------------------------------------
__global__ __launch_bounds__(256) void bn_relu_kernel(float* __restrict__ X,
                                                      const float* __restrict__ ssum,
                                                      const float* __restrict__ ssq,
                                                      const float* __restrict__ g,
                                                      const float* __restrict__ be) {
  const int c = blockIdx.y;
  const int p = blockIdx.x * 256 + threadIdx.x;
  const float inv  = 1.0f / (float)P_TOT;
  const float mean = ssum[c] * inv;
  const float var  = ssq[c] * inv - mean * mean;
  const float sc   = g[c] * rsqrtf(var + BN_EPS);
  const float sh   = be[c] - mean * sc;
  const size_t i = (size_t)c * P_TOT + p;
  float v = X[i] * sc + sh;
  X[i] = v > 0.f ? v : 0.f;
}

// ---------------------------------------------------------------------------
// 6) Max over K (=32 contiguous positions) -> new_points (B, S, 128).
// ---------------------------------------------------------------------------
__global__ __launch_bounds__(256) void max_kernel(const float* __restrict__ X,
                                                  float* __restrict__ out) {
  const int t  = blockIdx.x * 256 + threadIdx.x;
  const int c  = t & 127;
  const int bs = t >> 7;
  const float* x = X + (size_t)c * P_TOT + (size_t)bs * NSAMPLE_;
  float m = x[0];
#pragma unroll
  for (int k = 1; k < NSAMPLE_; ++k) m = fmaxf(m, x[k]);
  out[(size_t)bs * 128 + c] = m;
}

// ---------------------------------------------------------------------------
extern "C" void kernel_launch(void* const* d_in, const int* in_sizes, int n_in,
                              void* d_out, int out_size, void* d_ws, size_t ws_size,
                              hipStream_t stream) {
  const float* xyz = (const float*)d_in[0];
  const float* pts = (const float*)d_in[1];
  const float* w0 = (const float*)d_in[2];
  const float* b0 = (const float*)d_in[3];
  const float* g0 = (const float*)d_in[4];
  const float* be0 = (const float*)d_in[5];
  const float* w1 = (const float*)d_in[6];
  const float* b1 = (const float*)d_in[7];
  const float* g1 = (const float*)d_in[8];
  const float* be1 = (const float*)d_in[9];
  const float* w2 = (const float*)d_in[10];
  const float* b2 = (const float*)d_in[11];
  const float* g2 = (const float*)d_in[12];
  const float* be2 = (const float*)d_in[13];

  float* out        = (float*)d_out;
  float* new_xyz    = out;                         // B*NPOINT*3 = 49152 floats
  float* new_points = out + (size_t)B_ * NPOINT_ * 3;

  // workspace: bufA (64 rows), bufB (128 rows), stats (256 floats), Wpad (512)
  char*  ws   = (char*)d_ws;
  float* bufA = (float*)ws;
  float* bufB = (float*)(ws + (size_t)64 * P_TOT * sizeof(float));
  float* ssum = (float*)(ws + (size_t)(64 + 128) * P_TOT * sizeof(float));
  float* ssq  = ssum + 128;
  float* w0p  = ssq + 128;                         // 64*8 = 512 floats

  const int NT = P_TOT / 16;  // 32768 position tiles (waves cover 2 each)

  // 1) FPS -> new_xyz
  fps_kernel<<<B_, 1024, 0, stream>>>(xyz, new_xyz);

  // 2) ball query + grouping -> x0 (8 x P) in bufA
  group_kernel<<<(B_ * NPOINT_) / 8, 256, 0, stream>>>(xyz, pts, new_xyz, bufA);

  // pad layer-0 weights (6 -> 8 columns)
  pad_w_kernel<<<2, 256, 0, stream>>>(w0, w0p, 6, 8, 64 * 8);

  // 3) layer 0: (6->64)  X=bufA(8 rows) -> Y=bufB(64 rows)
  wmma_gemm_kernel<8><<<(4 * (NT / 2)) / 8, 256, 0, stream>>>(w0p, b0, bufA, bufB, 64);
  zero_stats_kernel<<<1, 256, 0, stream>>>(ssum, 256);
  stats_kernel<<<64 * 64, 256, 0, stream>>>(bufB, ssum, ssq, 64);
  bn_relu_kernel<<<dim3(P_TOT / 256, 64), 256, 0, stream>>>(bufB, ssum, ssq, g0, be0);

  // 4) layer 1: (64->64) X=bufB -> Y=bufA
  wmma_gemm_kernel<64><<<(4 * (NT / 2)) / 8, 256, 0, stream>>>(w1, b1, bufB, bufA, 64);
  zero_stats_kernel<<<1, 256, 0, stream>>>(ssum, 256);
  stats_kernel<<<64 * 64, 256, 0, stream>>>(bufA, ssum, ssq, 64);
  bn_relu_kernel<<<dim3(P_TOT / 256, 64), 256, 0, stream>>>(bufA, ssum, ssq, g1, be1);

  // 5) layer 2: (64->128) X=bufA -> Y=bufB
  wmma_gemm_kernel<64><<<(8 * (NT / 2)) / 8, 256, 0, stream>>>(w2, b2, bufA, bufB, 128);
  zero_stats_kernel<<<1, 256, 0, stream>>>(ssum, 256);
  stats_kernel<<<128 * 64, 256, 0, stream>>>(bufB, ssum, ssq, 64);
  bn_relu_kernel<<<dim3(P_TOT / 256, 128), 256, 0, stream>>>(bufB, ssum, ssq, g2, be2);

  // 6) max over K -> new_points
  max_kernel<<<(B_ * NPOINT_ * 128) / 256, 256, 0, stream>>>(bufB, new_points);
}